// AConvCircular2D_69028714381647
// MI455X (gfx1250) — compile-verified
//
#include <hip/hip_runtime.h>
#include <hip/hip_bf16.h>

typedef __attribute__((ext_vector_type(2))) float v2f;
typedef __attribute__((ext_vector_type(8))) float v8f;

#define NPOS 2304   // 48*48
#define HH 48

// ---------------------------------------------------------------------------
// one-shot weight transpose: OIHW (oc,ci,tap) -> (oc,tap,ci) so the implicit
// GEMM K-order is tap-major / ci-minor (contiguous staging, fixed tap/chunk).
// ---------------------------------------------------------------------------
__global__ __launch_bounds__(256) void wtrans_kernel(
    const float* __restrict__ w, float* __restrict__ wt, int total)
{
  int idx = blockIdx.x * 256 + threadIdx.x;
  if (idx >= total) return;
  int oc  = idx / 1152;
  int rem = idx - oc * 1152;
  int r   = rem >> 7;       // tap 0..8
  int ci  = rem & 127;      // 0..127
  wt[idx] = w[oc * 1152 + ci * 9 + r];
}

// ---------------------------------------------------------------------------
// conv3x3 circular, implicit GEMM: M=32 out-channels (2 tiles), N=16 positions,
// K=1152 ordered (tap, ci).  grid = (144, OC/32, 4), block = 32.
// Weight chunks staged to LDS with CDNA5 async global->LDS copies.
// ---------------------------------------------------------------------------
__global__ __launch_bounds__(32) void conv3x3_wmma_kernel(
    const float* __restrict__ x, const float* __restrict__ wt,
    float* __restrict__ out, int b_stride_out)
{
  const int tid = threadIdx.x;
  const int lo = tid & 15;
  const int hi = tid >> 4;
  const int nt = blockIdx.x;            // 0..143 position tile (16 in one row)
  const int oc0 = blockIdx.y * 32;
  const int b = blockIdx.z;
  const int y  = nt / 3;                // output row
  const int x0 = (nt % 3) * 16;         // first output col of tile

  __shared__ float wlds[32 * 68];       // 32 oc rows x 64 K (stride 68)

  const float* __restrict__ xb = x + b * (128 * NPOS);

  v8f acc0 = {}, acc1 = {};

  for (int c = 0; c < 18; ++c) {        // 18 chunks of K=64, tap fixed per chunk
    const int kb  = c * 64;
    const int r   = c >> 1;             // tap 0..8
    const int cib = (c & 1) * 64;       // ci base within the tap

    // make sure previous chunk's LDS reads are complete before async overwrite
    asm volatile("s_wait_dscnt 0x0" ::: "memory");
    // stage 32x64 weight chunk into LDS via async global->LDS (ASYNCcnt path)
    #pragma unroll
    for (int i = 0; i < 16; ++i) {
      int e   = i * 32 + tid;           // 0..511 float4 slots
      int row = e >> 4;                 // 0..31
      int c4  = e & 15;                 // 0..15
      const float* gp = wt + (oc0 + row) * 1152 + kb + c4 * 4;
      unsigned ldsa = (unsigned)(size_t)(&wlds[row * 68 + c4 * 4]);
      asm volatile("global_load_async_to_lds_b128 %0, %1, off"
                   :: "v"(ldsa), "v"(gp) : "memory");
    }
    asm volatile("s_wait_asynccnt 0x0" ::: "memory");

    // circular-wrap address math once per chunk (tap is fixed)
    const int ky = r / 3;
    const int kx = r - ky * 3;
    int yy = y + ky - 1;        yy += (yy < 0) ? HH : 0;  yy -= (yy >= HH) ? HH : 0;
    int xx = x0 + lo + kx - 1;  xx += (xx < 0) ? HH : 0;  xx -= (xx >= HH) ? HH : 0;
    const float* __restrict__ base = xb + ((size_t)cib * NPOS + yy * HH + xx);

    #pragma unroll 4
    for (int s = 0; s < 16; ++s) {
      v2f a0, a1, bf;
      #pragma unroll
      for (int v = 0; v < 2; ++v) {
        const int koff = s * 4 + v + 2 * hi;          // 0..63
        a0[v] = wlds[lo * 68 + koff];                 // A0[m=oc][k]
        a1[v] = wlds[(lo + 16) * 68 + koff];          // A1[m=oc+16][k]
        bf[v] = base[(size_t)koff * NPOS];            // B[k=ci][n=pos]
      }
      acc0 = __builtin_amdgcn_wmma_f32_16x16x4_f32(
          false, a0, false, bf, (short)0, acc0, false, false);
      acc1 = __builtin_amdgcn_wmma_f32_16x16x4_f32(
          false, a1, false, bf, (short)0, acc1, false, false);
    }
  }

  float* __restrict__ ob = out + (size_t)b * b_stride_out + oc0 * NPOS + nt * 16;
  #pragma unroll
  for (int v = 0; v < 8; ++v) {
    ob[(v + 8 * hi) * NPOS + lo] = acc0[v];
    ob[(16 + v + 8 * hi) * NPOS + lo] = acc1[v];
  }
}

// ---------------------------------------------------------------------------
// flash attention per (b, head, 16-query tile).  dk_h = dv_h = 16.
// 32 keys per iteration: S = Q*K^T (8 wmma), one exp2-domain online-softmax
// update, O^T = V^T * P^T (8 wmma).  grid = (144, 8, 4), block = 32.
// Output written in flat (b, h, n, dv) order == (b, ci, pos) for the 1x1 conv.
// ---------------------------------------------------------------------------
#define LOG2E 1.44269504088896340736f

__global__ __launch_bounds__(32) void attn_wmma_kernel(
    const float* __restrict__ qkv, float* __restrict__ attn_out)
{
  const int tid = threadIdx.x;
  const int lo = tid & 15;
  const int hi = tid >> 4;
  const int q0 = blockIdx.x * 16;
  const int h  = blockIdx.y;
  const int b  = blockIdx.z;

  const float* __restrict__ fq = qkv + (b * 384 +       h * 16) * NPOS;
  const float* __restrict__ fk = qkv + (b * 384 + 128 + h * 16) * NPOS;
  const float* __restrict__ fv = qkv + (b * 384 + 256 + h * 16) * NPOS;

  __shared__ float p_lds[16 * 33];   // 16 queries x 32 keys (stride 33)
  __shared__ float v_lds[16 * 33];   // 16 dv      x 32 keys (stride 33)
  __shared__ float c_lds[16];
  __shared__ float l_lds[16];

  // Q fragment (A: 16 queries x 16 dims), 1/sqrt(16) and log2(e) folded in
  v2f qa[4];
  #pragma unroll
  for (int s = 0; s < 4; ++s)
    #pragma unroll
    for (int v = 0; v < 2; ++v)
      qa[s][v] = fq[(s * 4 + v + 2 * hi) * NPOS + q0 + lo] * (0.25f * LOG2E);

  v8f o = {};
  float m_st[8], l_st[8];
  #pragma unroll
  for (int v = 0; v < 8; ++v) { m_st[v] = -__builtin_inff(); l_st[v] = 0.0f; }

  for (int kt = 0; kt < 72; ++kt) {            // 72 iterations x 32 keys
    const int k0 = kt * 32;
    if (kt + 1 < 72) {                         // prefetch next K/V tiles
      __builtin_prefetch(fk + k0 + 32, 0, 3);
      __builtin_prefetch(fv + k0 + 32, 0, 3);
    }

    // ---- S = Q * K^T for two 16-key tiles (log2 domain) --------------------
    v8f sf0 = {}, sf1 = {};
    #pragma unroll
    for (int s = 0; s < 4; ++s) {
      v2f bk0, bk1;
      #pragma unroll
      for (int v = 0; v < 2; ++v) {
        const int d = s * 4 + v + 2 * hi;                 // dim 0..15
        bk0[v] = fk[d * NPOS + k0 + lo];                  // B[k=dim][n=key]
        bk1[v] = fk[d * NPOS + k0 + 16 + lo];
      }
      sf0 = __builtin_amdgcn_wmma_f32_16x16x4_f32(
          false, qa[s], false, bk0, (short)0, sf0, false, false);
      sf1 = __builtin_amdgcn_wmma_f32_16x16x4_f32(
          false, qa[s], false, bk1, (short)0, sf1, false, false);
    }

    // ---- combined online softmax (rows at (vgpr, lane-half), cols in lanes)
    float mx[8];
    #pragma unroll
    for (int v = 0; v < 8; ++v) mx[v] = fmaxf(sf0[v], sf1[v]);
    #pragma unroll
    for (int msk = 1; msk <= 8; msk <<= 1)
      #pragma unroll
      for (int v = 0; v < 8; ++v)
        mx[v] = fmaxf(mx[v], __shfl_xor(mx[v], msk, 32));

    float pr0[8], pr1[8], rs[8], cr[8];
    #pragma unroll
    for (int v = 0; v < 8; ++v) {
      float mo = m_st[v];
      float mn = fmaxf(mo, mx[v]);
      cr[v] = __builtin_amdgcn_exp2f(mo - mn);            // base-2 correction
      m_st[v] = mn;
      pr0[v] = __builtin_amdgcn_exp2f(sf0[v] - mn);       // exp2 == exp here
      pr1[v] = __builtin_amdgcn_exp2f(sf1[v] - mn);
      rs[v] = pr0[v] + pr1[v];
    }
    #pragma unroll
    for (int msk = 1; msk <= 8; msk <<= 1)
      #pragma unroll
      for (int v = 0; v < 8; ++v)
        rs[v] += __shfl_xor(rs[v], msk, 32);
    #pragma unroll
    for (int v = 0; v < 8; ++v)
      l_st[v] = l_st[v] * cr[v] + rs[v];

    // broadcast per-query correction (row -> lane) through LDS
    #pragma unroll
    for (int v = 0; v < 8; ++v)
      if (lo == v) c_lds[hi * 8 + v] = cr[v];

    // P tile -> LDS (row = query, col = key 0..31)
    #pragma unroll
    for (int v = 0; v < 8; ++v) {
      p_lds[(v + 8 * hi) * 33 + lo] = pr0[v];
      p_lds[(v + 8 * hi) * 33 + 16 + lo] = pr1[v];
    }

    // V tile -> LDS coalesced: v_lds[dv][key 0..31]
    #pragma unroll
    for (int i = 0; i < 16; ++i) {
      int e = i * 32 + tid;            // 0..511
      int d = e >> 5, ke = e & 31;
      v_lds[d * 33 + ke] = fv[d * NPOS + k0 + ke];
    }

    const float cb = c_lds[lo];        // correction for this lane's query
    #pragma unroll
    for (int v = 0; v < 8; ++v) o[v] *= cb;

    // ---- O^T += V^T * P^T over 32 keys -------------------------------------
    #pragma unroll
    for (int s = 0; s < 8; ++s) {
      v2f av, bp;
      #pragma unroll
      for (int v = 0; v < 2; ++v) {
        const int idx = s * 4 + v + 2 * hi;      // key index 0..31
        av[v] = v_lds[lo * 33 + idx];            // A[m=dv][k=key]
        bp[v] = p_lds[lo * 33 + idx];            // B[k=key][n=query]
      }
      o = __builtin_amdgcn_wmma_f32_16x16x4_f32(
          false, av, false, bp, (short)0, o, false, false);
    }
  }

  // broadcast row sums (row -> lane), normalize, store flat (b,h,n,dv)
  #pragma unroll
  for (int v = 0; v < 8; ++v)
    if (lo == v) l_lds[hi * 8 + v] = l_st[v];
  const float inv = 1.0f / l_lds[lo];

  float* __restrict__ ob = attn_out + ((size_t)(b * 8 + h) * NPOS + q0) * 16;
  #pragma unroll
  for (int v = 0; v < 8; ++v)
    ob[(size_t)lo * 16 + (v + 8 * hi)] = o[v] * inv;   // O[q=lo][dv=v+8hi]
}

// ---------------------------------------------------------------------------
// 1x1 conv: out2[b][oc][p] = sum_ci w_out[oc][ci] * attn[b][ci][p]
// grid = (144, 8, 4), block = 32. K=128 -> 32 wmma steps.
// ---------------------------------------------------------------------------
__global__ __launch_bounds__(32) void conv1x1_wmma_kernel(
    const float* __restrict__ attn, const float* __restrict__ wout,
    float* __restrict__ out)
{
  const int tid = threadIdx.x;
  const int lo = tid & 15;
  const int hi = tid >> 4;
  const int nt = blockIdx.x;
  const int oc0 = blockIdx.y * 16;
  const int b = blockIdx.z;

  __shared__ float wlds[16 * 132];      // 16 oc x 128 ci, stride 132 (f4-aligned)

  #pragma unroll
  for (int i = 0; i < 16; ++i) {
    int e   = i * 32 + tid;             // 0..511 float4 slots
    int row = e >> 5;                   // 0..15
    int c4  = e & 31;                   // 0..31
    const float4* src = reinterpret_cast<const float4*>(wout + (oc0 + row) * 128 + c4 * 4);
    *reinterpret_cast<float4*>(&wlds[row * 132 + c4 * 4]) = *src;
  }

  const float* __restrict__ ab = attn + (size_t)b * (128 * NPOS) + nt * 16;
  v8f acc = {};
  #pragma unroll 4
  for (int s = 0; s < 32; ++s) {
    v2f a, bf;
    #pragma unroll
    for (int v = 0; v < 2; ++v) {
      const int k = s * 4 + v + 2 * hi;          // ci 0..127
      a[v]  = wlds[lo * 132 + k];                // A[m=oc][k=ci]
      bf[v] = ab[(size_t)k * NPOS + lo];         // B[k=ci][n=pos]
    }
    acc = __builtin_amdgcn_wmma_f32_16x16x4_f32(
        false, a, false, bf, (short)0, acc, false, false);
  }

  float* __restrict__ ob = out + ((size_t)b * 256 + 128 + oc0) * NPOS + nt * 16;
  #pragma unroll
  for (int v = 0; v < 8; ++v)
    ob[(v + 8 * hi) * NPOS + lo] = acc[v];
}

// ---------------------------------------------------------------------------
extern "C" void kernel_launch(void* const* d_in, const int* in_sizes, int n_in,
                              void* d_out, int out_size, void* d_ws, size_t ws_size,
                              hipStream_t stream) {
  const float* x      = (const float*)d_in[0];   // (4,128,48,48)
  const float* w_init = (const float*)d_in[1];   // (128,128,3,3)
  const float* w_qkv  = (const float*)d_in[2];   // (384,128,3,3)
  const float* w_out  = (const float*)d_in[3];   // (128,128,1,1)
  float* out = (float*)d_out;                    // (4,256,48,48)

  float* qkv    = (float*)d_ws;                        // 4*384*2304 floats
  float* attn   = qkv  + (size_t)4 * 384 * NPOS;       // 4*128*2304 floats
  float* wi_t   = attn + (size_t)4 * 128 * NPOS;       // 128*1152 floats
  float* wq_t   = wi_t + (size_t)128 * 1152;           // 384*1152 floats

  // one-shot weight transposes into workspace
  wtrans_kernel<<<(128 * 1152 + 255) / 256, 256, 0, stream>>>(w_init, wi_t, 128 * 1152);
  wtrans_kernel<<<(384 * 1152 + 255) / 256, 256, 0, stream>>>(w_qkv,  wq_t, 384 * 1152);

  // channels 0..127 of output: init conv
  conv3x3_wmma_kernel<<<dim3(144, 4, 4), 32, 0, stream>>>(x, wi_t, out, 256 * NPOS);
  // qkv conv into workspace
  conv3x3_wmma_kernel<<<dim3(144, 12, 4), 32, 0, stream>>>(x, wq_t, qkv, 384 * NPOS);
  // attention into workspace (flat (b,h,n,dv) == (b,ci,pos))
  attn_wmma_kernel<<<dim3(144, 8, 4), 32, 0, stream>>>(qkv, attn);
  // channels 128..255 of output: 1x1 conv
  conv1x1_wmma_kernel<<<dim3(144, 8, 4), 32, 0, stream>>>(attn, w_out, out);
}